// GaussianMomentDescriptorT_66838281060517
// MI455X (gfx1250) — compile-verified
//
#include <hip/hip_runtime.h>
#include <hip/hip_bf16.h>

#define N_RADIAL 5
#define N_BASIS 7
#define R_MAXF 6.0f
#define N_ATOMS 10000
#define N_EDGES 600000
#define N_SPECIES 10
#define CPA 100          // compact moment floats per atom: m0(5)+m1(15)+m2c(30)+m3c(50)
#define NFEAT 360
#define PI_F 3.14159265358979323846f

typedef __attribute__((ext_vector_type(2))) float v2f;
typedef __attribute__((ext_vector_type(8))) float v8f;

// map full m2 index (i*3+j) -> compact sorted-pair index
__device__ __constant__ int c2map[9]  = {0,1,2, 1,3,4, 2,4,5};
// map full m3 index (i*9+j*3+k) -> compact sorted-triple index
__device__ __constant__ int c3map[27] = {0,1,2, 1,3,4, 2,4,5,
                                         1,3,4, 3,6,7, 4,7,8,
                                         2,4,5, 4,7,8, 5,8,9};
// tril_2d(5): 15 pairs
__device__ __constant__ int T2I[15] = {0,1,1,2,2,2,3,3,3,3,4,4,4,4,4};
__device__ __constant__ int T2J[15] = {0,0,1,0,1,2,0,1,2,3,0,1,2,3,4};
// tril_3d(5): 35 triples
__device__ __constant__ int T3I[35] = {0, 1,1,1, 2,2,2,2,2,2,
                                       3,3,3,3,3,3,3,3,3,3,
                                       4,4,4,4,4,4,4,4,4,4,4,4,4,4,4};
__device__ __constant__ int T3J[35] = {0, 0,1,1, 0,1,1,2,2,2,
                                       0,1,1,2,2,2,3,3,3,3,
                                       0,1,1,2,2,2,3,3,3,3,4,4,4,4,4};
__device__ __constant__ int T3K[35] = {0, 0,0,1, 0,0,1,0,1,2,
                                       0,0,1,0,1,2,0,1,2,3,
                                       0,0,1,0,1,2,0,1,2,3,0,1,2,3,4};

// per-wave LDS layout (floats)
#define L_CM   0      // compact moments, 100 (+4 pad)
#define L_XP   104    // Xp[16][28] = 448  (rows 0-4 m1, 5-9 m2, 10-14 m3; zero padded)
#define L_YP   552    // Yp[16][12] = 192  (Y[(r*3+k)][ij] = m3[r,i,j,k])
#define L_ZP   744    // Zp[16][4]  = 64   (Z[(r*3+j)][i] = m2[r,i,j])
#define L_BT   808    // Bt[16][12] = 192  (Bt[n][k] = m2full[n*9+k], n<5)
#define L_G1   1000   // 256
#define L_GC6  1256   // 256
#define L_BC4  1512   // 256
#define L_V7   1768   // 256
#define WAVE_LDS 2024
#define PAD_FLOATS 896   // Xp..Bt contiguous zero region

__global__ void gm_zero_kernel(float4* __restrict__ ws, int n4) {
  int i = blockIdx.x * blockDim.x + threadIdx.x;
  if (i < n4) ws[i] = make_float4(0.f, 0.f, 0.f, 0.f);
}

// One thread per edge: compute radial channels, accumulate compact symmetric
// moments into ws via f32 atomics (100 per contributing edge).
__global__ void gm_edge_kernel(const float* __restrict__ dr_vec,
                               const int*   __restrict__ Z,
                               const int*   __restrict__ nbr,
                               const float* __restrict__ W,
                               float*       __restrict__ acc) {
  int e = blockIdx.x * blockDim.x + threadIdx.x;
  if (e >= N_EDGES) return;
  float dx = dr_vec[3*e+0], dy = dr_vec[3*e+1], dz = dr_vec[3*e+2];
  float dr = sqrtf(dx*dx + dy*dy + dz*dz);
  if (dr >= R_MAXF) return;                 // cutoff==0 -> exact-zero contribution
  float inv = 1.0f / (dr + 1e-5f);
  float dn0 = dx*inv, dn1 = dy*inv, dn2 = dz*inv;

  int ai = nbr[e];
  int aj = nbr[N_EDGES + e];
  int zi = Z[ai], zj = Z[aj];
  const float* w = W + ((zi*N_SPECIES + zj) * N_RADIAL * N_BASIS);

  float cutoff = 0.5f * (cosf(PI_F * dr / R_MAXF) + 1.0f);

  const float betta = (7.0f/6.0f)*(7.0f/6.0f);
  float basis[N_BASIS];
  #pragma unroll
  for (int k = 0; k < N_BASIS; ++k) {
    float d = dr - (float)k;                // shifts = 0,1,...,6
    basis[k] = expf(-betta * d * d);
  }
  float rad[N_RADIAL];
  #pragma unroll
  for (int r = 0; r < N_RADIAL; ++r) {
    float s = 0.0f;
    #pragma unroll
    for (int k = 0; k < N_BASIS; ++k) s += basis[k] * w[r*N_BASIS + k];
    rad[r] = cutoff * s;
  }

  float p2[6]  = { dn0*dn0, dn0*dn1, dn0*dn2, dn1*dn1, dn1*dn2, dn2*dn2 };
  float p3[10] = { p2[0]*dn0, p2[0]*dn1, p2[0]*dn2, p2[3]*dn0, p2[1]*dn2,
                   p2[5]*dn0, p2[3]*dn1, p2[3]*dn2, p2[5]*dn1, p2[5]*dn2 };
  float dn[3] = { dn0, dn1, dn2 };

  float* base = acc + (size_t)aj * CPA;     // segment by idx_j
  #pragma unroll
  for (int r = 0; r < N_RADIAL; ++r) {
    float f0 = rad[r];
    atomicAdd(base + r, f0);
    #pragma unroll
    for (int i = 0; i < 3; ++i)  atomicAdd(base + 5  + r*3  + i, f0 * dn[i]);
    #pragma unroll
    for (int s = 0; s < 6; ++s)  atomicAdd(base + 20 + r*6  + s, f0 * p2[s]);
    #pragma unroll
    for (int s = 0; s < 10; ++s) atomicAdd(base + 50 + r*10 + s, f0 * p3[s]);
  }
}

union U8 { v8f v; float f[8]; };

// One atom per wave32; 4 waves per block. WMMA f32 16x16x4 Gram contractions
// with branchless ds_load_b64 operand fetch from zero-padded LDS matrices.
__global__ void __launch_bounds__(128)
gm_contract_kernel(const float* __restrict__ acc, float* __restrict__ out) {
  __shared__ __align__(16) float lds[4][WAVE_LDS];
  const int wave = threadIdx.x >> 5;
  const int lane = threadIdx.x & 31;
  const int atom = blockIdx.x * 4 + wave;

  float* cm  = lds[wave] + L_CM;
  float* Xp  = lds[wave] + L_XP;
  float* Yp  = lds[wave] + L_YP;
  float* Zp  = lds[wave] + L_ZP;
  float* Bt  = lds[wave] + L_BT;
  float* G1  = lds[wave] + L_G1;
  float* GC6 = lds[wave] + L_GC6;
  float* BC4 = lds[wave] + L_BC4;
  float* V7  = lds[wave] + L_V7;

  // stage compact moments + zero the padded operand region
  const float* c = acc + (size_t)atom * CPA;
  for (int t = lane; t < CPA; t += 32) cm[t] = c[t];
  for (int t = lane; t < PAD_FLOATS; t += 32) Xp[t] = 0.0f;
  __syncthreads();

  // fill padded operand matrices from compact moments (branchy, but one-time)
  for (int e = lane; e < 195; e += 32) {          // Xp valid entries
    int row, k; float v;
    if (e < 15)      { row = e/3;          k = e%3;  v = cm[5 + e]; }
    else if (e < 60) { int q = e-15; row = 5 + q/9;  k = q%9;
                       v = cm[20 + (q/9)*6  + c2map[q%9]]; }
    else             { int q = e-60; row = 10 + q/27; k = q%27;
                       v = cm[50 + (q/27)*10 + c3map[q%27]]; }
    Xp[row*28 + k] = v;
  }
  for (int e = lane; e < 135; e += 32) {          // Yp[(r*3+k)][ij] = m3[r,i,j,k]
    int p = e/9, q = e%9;
    Yp[p*12 + q] = cm[50 + (p/3)*10 + c3map[q*3 + (p%3)]];
  }
  for (int e = lane; e < 45; e += 32) {           // Zp[(r*3+j)][i] = m2[r,i,j]
    int p = e/3, i = e%3;
    Zp[p*4 + i] = cm[20 + (p/3)*6 + c2map[i*3 + (p%3)]];
  }
  for (int e = lane; e < 45; e += 32) {           // Bt[n][k] = m2full[n*9+k]
    int n = e/9, k = e%9;
    Bt[n*12 + k] = cm[20 + n*6 + c2map[k]];
  }
  __syncthreads();

  const int m  = lane & 15;
  const int hi = lane >> 4;
  const int ko = hi * 2;                          // lane's K sub-offset

  // --- Gram1: X(15x28) X^T -> c1,c2,c3 (A and B registers coincide) ---
  {
    v8f g = {0.f,0.f,0.f,0.f,0.f,0.f,0.f,0.f};
    #pragma unroll
    for (int kk = 0; kk < 7; ++kk) {
      v2f a = *(const v2f*)(Xp + m*28 + kk*4 + ko);
      g = __builtin_amdgcn_wmma_f32_16x16x4_f32(false, a, false, a, (short)0, g, false, false);
    }
    U8 u; u.v = g;
    #pragma unroll
    for (int v = 0; v < 8; ++v) G1[(v + hi*8)*16 + m] = u.f[v];
  }
  // --- Gram Y(15x9) Y^T for c6 kernel ---
  {
    v8f g = {0.f,0.f,0.f,0.f,0.f,0.f,0.f,0.f};
    #pragma unroll
    for (int kk = 0; kk < 3; ++kk) {
      v2f a = *(const v2f*)(Yp + m*12 + kk*4 + ko);
      g = __builtin_amdgcn_wmma_f32_16x16x4_f32(false, a, false, a, (short)0, g, false, false);
    }
    U8 u; u.v = g;
    #pragma unroll
    for (int v = 0; v < 8; ++v) GC6[(v + hi*8)*16 + m] = u.f[v];
  }
  // --- Gram Z(15x3) Z^T for c4 kernel ---
  {
    v2f a = *(const v2f*)(Zp + m*4 + ko);
    v8f g = {0.f,0.f,0.f,0.f,0.f,0.f,0.f,0.f};
    g = __builtin_amdgcn_wmma_f32_16x16x4_f32(false, a, false, a, (short)0, g, false, false);
    U8 u; u.v = g;
    #pragma unroll
    for (int v = 0; v < 8; ++v) BC4[(v + hi*8)*16 + m] = u.f[v];
  }
  // --- V7 = Y(15x9) @ m2full^T(9x5) for c7 kernel ---
  {
    v8f g = {0.f,0.f,0.f,0.f,0.f,0.f,0.f,0.f};
    #pragma unroll
    for (int kk = 0; kk < 3; ++kk) {
      v2f a = *(const v2f*)(Yp + m*12 + kk*4 + ko);
      v2f b = *(const v2f*)(Bt + m*12 + kk*4 + ko);
      g = __builtin_amdgcn_wmma_f32_16x16x4_f32(false, a, false, b, (short)0, g, false, false);
    }
    U8 u; u.v = g;
    #pragma unroll
    for (int v = 0; v < 8; ++v) V7[(v + hi*8)*16 + m] = u.f[v];
  }
  __syncthreads();

  // --- assemble 360 features per atom ---
  float* orow = out + (size_t)atom * NFEAT;
  for (int f = lane; f < NFEAT; f += 32) {
    float val;
    if (f < 5) {
      val = cm[f];                               // m0
    } else if (f < 20) {                         // c1
      int p = f - 5;  val = G1[T2I[p]*16 + T2J[p]];
    } else if (f < 35) {                         // c2
      int p = f - 20; val = G1[(5 + T2I[p])*16 + 5 + T2J[p]];
    } else if (f < 50) {                         // c3
      int p = f - 35; val = G1[(10 + T2I[p])*16 + 10 + T2J[p]];
    } else if (f < 85) {                         // c4
      int p = f - 50; int r = T3I[p], s = T3J[p], t = T3K[p];
      float sum = 0.0f;
      #pragma unroll
      for (int jj = 0; jj < 3; ++jj)
        #pragma unroll
        for (int kk = 0; kk < 3; ++kk)
          sum += BC4[(r*3+jj)*16 + (s*3+kk)] * Xp[(5+t)*28 + jj*3 + kk];
      val = sum;
    } else if (f < 160) {                        // c5
      int p = f - 85; int pi = p / 5, t = p % 5;
      int r = T2I[pi], s = T2J[pi];
      float sum = 0.0f;
      #pragma unroll
      for (int i = 0; i < 3; ++i)
        #pragma unroll
        for (int j = 0; j < 3; ++j)
          sum += Xp[r*28 + i] * Xp[s*28 + j] * Xp[(5+t)*28 + i*3 + j];
      val = sum;
    } else if (f < 235) {                        // c6
      int p = f - 160; int pi = p / 5, t = p % 5;
      int r = T2I[pi], s = T2J[pi];
      float sum = 0.0f;
      #pragma unroll
      for (int k = 0; k < 3; ++k)
        #pragma unroll
        for (int l = 0; l < 3; ++l)
          sum += GC6[(r*3+k)*16 + (s*3+l)] * Xp[(5+t)*28 + k*3 + l];
      val = sum;
    } else {                                     // c7
      int p = f - 235; int r = p / 25, s = (p / 5) % 5, t = p % 5;
      float sum = 0.0f;
      #pragma unroll
      for (int k = 0; k < 3; ++k)
        sum += V7[(r*3+k)*16 + s] * Xp[t*28 + k];
      val = sum;
    }
    orow[f] = val;
  }
}

extern "C" void kernel_launch(void* const* d_in, const int* in_sizes, int n_in,
                              void* d_out, int out_size, void* d_ws, size_t ws_size,
                              hipStream_t stream) {
  const float* dr_vec = (const float*)d_in[0];
  const int*   Z      = (const int*)  d_in[1];
  const int*   nbr    = (const int*)  d_in[2];
  const float* W      = (const float*)d_in[3];
  float* out = (float*)d_out;
  float* ws  = (float*)d_ws;   // needs N_ATOMS*CPA*4 = 4 MB

  int n4 = (N_ATOMS * CPA) / 4;
  gm_zero_kernel<<<(n4 + 255) / 256, 256, 0, stream>>>((float4*)ws, n4);
  gm_edge_kernel<<<(N_EDGES + 255) / 256, 256, 0, stream>>>(dr_vec, Z, nbr, W, ws);
  gm_contract_kernel<<<N_ATOMS / 4, 128, 0, stream>>>(ws, out);
}